// PlanetWarsAgentGNN_25537875542624
// MI455X (gfx1250) — compile-verified
//
#include <hip/hip_runtime.h>

// ---------------------------------------------------------------------------
// GATv2 GNN agent forward for MI455X (gfx1250, wave32).
// Dense projections use V_WMMA_F32_16X16X4_F32 (fp32 matrix pipe).
// Edge attention / segment softmax / scatter are wave32 kernels with
// float global atomics; edge kernels are 2-way software-pipelined with
// global_prefetch (__builtin_prefetch) to hide L2 gather latency —
// the xl/xr tables (<=41 MB) live in the 192 MB L2, so the gather phase
// is L2-latency bound, not HBM bound.
// ---------------------------------------------------------------------------

typedef __attribute__((ext_vector_type(2))) float v2f;
typedef __attribute__((ext_vector_type(8))) float v8f;

#define N_NODES 20000
#define N_EDGES 320000
#define NGRAPH  32

// ---------------------------------------------------------------------------
// Y[rows,M] = X[rows,K] @ W[M,K]^T + bias, optional ReLU.
// One wave computes one 16x16 tile via V_WMMA_F32_16X16X4_F32.
// A layout: lanes 0-15 -> rows, VGPR pair holds K=0/1 (K=2/3 on lanes 16-31).
// B layout symmetric (B = W^T, so B[k][n] = W[n][k]).
// Requires rows % 16 == 0 and M % 64 == 0 (true for all call sites),
// so EXEC is all-ones at every WMMA (ISA requirement).
// ---------------------------------------------------------------------------
__global__ void gemm_bias_wmma(const float* __restrict__ X,
                               const float* __restrict__ W,
                               const float* __restrict__ bias,
                               float* __restrict__ Y,
                               int K, int M, int relu) {
  const int wave = threadIdx.x >> 5;     // 4 waves/block, 16 cols each
  const int lane = threadIdx.x & 31;
  const int colb = blockIdx.x * 64 + wave * 16;
  const int rowb = blockIdx.y * 16;
  const int l15  = lane & 15;
  const int khalf = (lane >> 4) * 2;     // 0 for lanes 0-15, 2 for 16-31

  const float* xrow = X + (size_t)(rowb + l15) * K + khalf;
  const float* wrow = W + (size_t)(colb + l15) * K + khalf;

  v8f acc = {};
  for (int k = 0; k < K; k += 4) {
    v2f a; a.x = xrow[k]; a.y = xrow[k + 1];
    v2f b; b.x = wrow[k]; b.y = wrow[k + 1];
    acc = __builtin_amdgcn_wmma_f32_16x16x4_f32(
        /*neg_a=*/false, a, /*neg_b=*/false, b,
        /*c_mod=*/(short)0, acc, /*reuse_a=*/false, /*reuse_b=*/false);
  }

  const int   col = colb + l15;
  const float bv  = bias[col];
#pragma unroll
  for (int v = 0; v < 8; ++v) {
    const int row = rowb + ((lane < 16) ? v : v + 8);
    float r = acc[v] + bv;
    if (relu) r = fmaxf(r, 0.0f);
    Y[(size_t)row * M + col] = r;
  }
}

// ---------------------------------------------------------------------------
// utility fills
// ---------------------------------------------------------------------------
__global__ void fill_kernel(float* p, float v, int n) {
  int i = blockIdx.x * blockDim.x + threadIdx.x;
  if (i < n) p[i] = v;
}

__global__ void fill_bias_rows(float* p, const float* __restrict__ bias,
                               int rows, int dim) {
  int i = blockIdx.x * blockDim.x + threadIdx.x;
  if (i < rows * dim) p[i] = bias[i % dim];   // dim is a power of two
}

// ---------------------------------------------------------------------------
// Edge attention: alpha[e,h] = att_h . leaky_relu(xl[src] + xr[dst] + We@ea)
// One wave per 2 edges (e, e+E/2); We/att staged in LDS; the second edge's
// gather rows are prefetched (global_prefetch) before computing the first,
// hiding L2 gather latency behind ALU work.
// ---------------------------------------------------------------------------
template <int H, int C>
__global__ void edge_att_kernel(const float* __restrict__ xl,
                                const float* __restrict__ xr,
                                const float* __restrict__ ea,
                                const int* __restrict__ ei,
                                const float* __restrict__ We,
                                const float* __restrict__ att,
                                float* __restrict__ alpha, int E) {
  constexpr int HC  = H * C;
  constexpr int CPL = HC / 32;     // channels per lane
  constexpr int GRP = 32 / H;      // lanes per head
  __shared__ float sWe[HC * 5];
  __shared__ float sAtt[HC];
  for (int i = threadIdx.x; i < HC * 5; i += blockDim.x) sWe[i] = We[i];
  for (int i = threadIdx.x; i < HC; i += blockDim.x) sAtt[i] = att[i];
  __syncthreads();

  const int wave = threadIdx.x >> 5;
  const int lane = threadIdx.x & 31;
  const int EH = E >> 1;                       // E is even
  const int e0 = blockIdx.x * (blockDim.x >> 5) + wave;
  if (e0 >= EH) return;
  const int e1 = e0 + EH;

  const int src0 = ei[e0], dst0 = ei[E + e0];
  const int src1 = ei[e1], dst1 = ei[E + e1];
  const int c0 = lane * CPL;

  // prefetch second edge's gather rows (lowered to global_prefetch_b8)
  __builtin_prefetch(xl + (size_t)src1 * HC + c0, 0, 3);
  __builtin_prefetch(xr + (size_t)dst1 * HC + c0, 0, 3);

  auto do_edge = [&](int e, int src, int dst) {
    float ed[5];
#pragma unroll
    for (int d = 0; d < 5; ++d) ed[d] = ea[(size_t)e * 5 + d];
    const float* pl = xl + (size_t)src * HC + c0;
    const float* pr = xr + (size_t)dst * HC + c0;
    float part = 0.0f;
#pragma unroll
    for (int c = 0; c < CPL; ++c) {
      float v = pl[c] + pr[c];
      const float* w = &sWe[(c0 + c) * 5];
#pragma unroll
      for (int d = 0; d < 5; ++d) v = fmaf(ed[d], w[d], v);
      v = (v > 0.0f) ? v : 0.2f * v;           // leaky_relu, slope 0.2
      part = fmaf(v, sAtt[c0 + c], part);
    }
#pragma unroll
    for (int off = GRP >> 1; off > 0; off >>= 1)
      part += __shfl_xor(part, off, 32);
    if ((lane & (GRP - 1)) == 0)
      alpha[(size_t)e * H + (lane / GRP)] = part;
  };

  do_edge(e0, src0, dst0);
  do_edge(e1, src1, dst1);
}

// ---------------------------------------------------------------------------
// segment softmax pieces
// ---------------------------------------------------------------------------
__device__ inline void atomicMaxF(float* addr, float val) {
  int* ia = (int*)addr;
  int old = __float_as_int(*addr);
  while (__int_as_float(old) < val) {
    int assumed = old;
    old = atomicCAS(ia, assumed, __float_as_int(val));
    if (old == assumed) break;
  }
}

__global__ void seg_max_kernel(const float* __restrict__ alpha,
                               const int* __restrict__ ei,
                               float* m, int E, int H) {
  int i = blockIdx.x * blockDim.x + threadIdx.x;
  if (i >= E * H) return;
  const int e = i / H, h = i - e * H;
  atomicMaxF(&m[ei[E + e] * H + h], alpha[i]);
}

__global__ void exp_sum_kernel(float* alpha, const int* __restrict__ ei,
                               const float* __restrict__ m, float* d,
                               int E, int H) {
  int i = blockIdx.x * blockDim.x + threadIdx.x;
  if (i >= E * H) return;
  const int e = i / H, h = i - e * H;
  const int dst = ei[E + e];
  const float a = __expf(alpha[i] - m[dst * H + h]);
  alpha[i] = a;
  atomicAdd(&d[dst * H + h], a);
}

// out[dst] += xl[src] * alpha / (d[dst] + eps); one wave per 2 edges,
// with prefetch of the second edge's source row.
template <int H, int C>
__global__ void scatter_kernel(const float* __restrict__ xl,
                               const float* __restrict__ alpha,
                               const float* __restrict__ dsum,
                               const int* __restrict__ ei,
                               float* out, int E) {
  constexpr int HC  = H * C;
  constexpr int CPL = HC / 32;
  constexpr int GRP = 32 / H;
  const int wave = threadIdx.x >> 5;
  const int lane = threadIdx.x & 31;
  const int EH = E >> 1;
  const int e0 = blockIdx.x * (blockDim.x >> 5) + wave;
  if (e0 >= EH) return;
  const int e1 = e0 + EH;

  const int src0 = ei[e0], dst0 = ei[E + e0];
  const int src1 = ei[e1], dst1 = ei[E + e1];
  const int coff = lane * CPL;
  const int h = lane / GRP;

  __builtin_prefetch(xl + (size_t)src1 * HC + coff, 0, 3);

  auto do_edge = [&](int e, int src, int dst) {
    const float coef =
        alpha[(size_t)e * H + h] / (dsum[dst * H + h] + 1e-16f);
    const float* pl = xl + (size_t)src * HC + coff;
    float* po = out + (size_t)dst * HC + coff;
#pragma unroll
    for (int c = 0; c < CPL; ++c) atomicAdd(&po[c], pl[c] * coef);
  };

  do_edge(e0, src0, dst0);
  do_edge(e1, src1, dst1);
}

// ---------------------------------------------------------------------------
// segment mean over batch (one wave per node), finalize, mean-sub + relu
// ---------------------------------------------------------------------------
__global__ void seg_sum_kernel(const float* __restrict__ h,
                               const int* __restrict__ batch,
                               float* sums, float* cnt, int N, int D) {
  const int wave = threadIdx.x >> 5;
  const int lane = threadIdx.x & 31;
  const int n = blockIdx.x * (blockDim.x >> 5) + wave;
  if (n >= N) return;
  const int b = batch[n];
  for (int c = lane; c < D; c += 32)
    atomicAdd(&sums[b * D + c], h[(size_t)n * D + c]);
  if (lane == 0) atomicAdd(&cnt[b], 1.0f);
}

__global__ void finalize_mean(float* sums, const float* __restrict__ cnt,
                              int B, int D) {
  int i = blockIdx.x * blockDim.x + threadIdx.x;
  if (i >= B * D) return;
  sums[i] /= fmaxf(cnt[i / D], 1.0f);
}

__global__ void sub_mean_relu(float* h, const float* __restrict__ mean,
                              const int* __restrict__ batch, int N, int D) {
  int i = blockIdx.x * blockDim.x + threadIdx.x;
  if (i >= N * D) return;
  const int n = i / D;
  const int c = i - n * D;
  h[i] = fmaxf(h[i] - mean[batch[n] * D + c], 0.0f);
}

// MLP output layer (out dim == 1): one wave per row.
__global__ void dot_rows_kernel(const float* __restrict__ X,
                                const float* __restrict__ w,
                                const float* __restrict__ b2,
                                float* out, int N, int K) {
  const int wave = threadIdx.x >> 5;
  const int lane = threadIdx.x & 31;
  const int n = blockIdx.x * (blockDim.x >> 5) + wave;
  if (n >= N) return;
  float s = 0.0f;
  for (int c = lane; c < K; c += 32) s = fmaf(X[(size_t)n * K + c], w[c], s);
#pragma unroll
  for (int off = 16; off > 0; off >>= 1) s += __shfl_xor(s, off, 32);
  if (lane == 0) out[n] = s + b2[0];
}

// ---------------------------------------------------------------------------
// host orchestration
// ---------------------------------------------------------------------------
struct Gat { const float *We, *Wl, *Wr, *att, *bias, *bl, *br; };
struct Mlp { const float *W1, *W2, *b1, *b2; };

static void gemm(hipStream_t s, const float* X, const float* W,
                 const float* b, float* Y, int rows, int K, int M, int relu) {
  dim3 g(M / 64, rows / 16), blk(128);
  gemm_bias_wmma<<<g, blk, 0, s>>>(X, W, b, Y, K, M, relu);
}

extern "C" void kernel_launch(void* const* d_in, const int* in_sizes, int n_in,
                              void* d_out, int out_size, void* d_ws,
                              size_t ws_size, hipStream_t stream) {
  const float* x     = (const float*)d_in[0];
  const int*   ei    = (const int*)d_in[1];
  const float* ea    = (const float*)d_in[2];
  const int*   batch = (const int*)d_in[3];
  auto F = [&](int i) { return (const float*)d_in[i]; };
  // jax pytree (sorted-key) flattening: a1,a2,critic,noop,source,v1,v2;
  // GAT: We,Wl,Wr,att,bias,bl,br ; MLP: W1,W2,b1,b2
  Gat a1{F(4),  F(5),  F(6),  F(7),  F(8),  F(9),  F(10)};
  Gat a2{F(11), F(12), F(13), F(14), F(15), F(16), F(17)};
  Mlp critic{F(18), F(19), F(20), F(21)};
  Mlp noopm {F(22), F(23), F(24), F(25)};
  Mlp srcm  {F(26), F(27), F(28), F(29)};
  Gat v1{F(30), F(31), F(32), F(33), F(34), F(35), F(36)};
  Gat v2{F(37), F(38), F(39), F(40), F(41), F(42), F(43)};

  // output layout (return order, flat)
  float* out_h     = (float*)d_out;                 // [N,128]
  float* out_g     = out_h + (size_t)N_NODES * 128; // [32,128]
  float* out_value = out_g + NGRAPH * 128;          // [32,1]
  float* out_src   = out_value + NGRAPH;            // [N]
  float* out_noop  = out_src + N_NODES;             // [32,1]

  // workspace layout (floats), peak ~140 MB
  float* W    = (float*)d_ws;
  float* xl1  = W;                               // N*512 (layer2 reuses as xl2)
  float* xr1  = xl1 + (size_t)N_NODES * 512;     // N*512
  float* h1   = xr1 + (size_t)N_NODES * 512;     // N*512
  float* alp  = h1  + (size_t)N_NODES * 512;     // E*4 (layer2 reuses E*1)
  float* mbuf = alp + (size_t)N_EDGES * 4;       // N*4
  float* dbuf = mbuf + (size_t)N_NODES * 4;      // N*4
  float* meanb= dbuf + (size_t)N_NODES * 4;      // 32*512
  float* cntb = meanb + NGRAPH * 512;            // 32
  float* hv   = cntb + NGRAPH;                   // N*128 (value h2; src hidden)
  float* gv   = hv + (size_t)N_NODES * 128;      // 32*128
  float* hid32= gv + NGRAPH * 128;               // 32*128

  const int EW2 = (N_EDGES / 2 + 7) / 8;         // 2 edges/wave, 8 waves/blk

  auto run_gnn = [&](const Gat& p1, const Gat& p2, float* hout) {
    // ---- layer 1: H=4, C=128 (HC=512), in_dim=8 ----
    gemm(stream, x, p1.Wl, p1.bl, xl1, N_NODES, 8, 512, 0);
    gemm(stream, x, p1.Wr, p1.br, xr1, N_NODES, 8, 512, 0);
    edge_att_kernel<4,128><<<EW2, 256, 0, stream>>>(xl1, xr1, ea, ei,
                                                    p1.We, p1.att, alp, N_EDGES);
    fill_kernel<<<(N_NODES*4+255)/256, 256, 0, stream>>>(mbuf, -3.0e38f, N_NODES*4);
    seg_max_kernel<<<(N_EDGES*4+255)/256, 256, 0, stream>>>(alp, ei, mbuf, N_EDGES, 4);
    fill_kernel<<<(N_NODES*4+255)/256, 256, 0, stream>>>(dbuf, 0.0f, N_NODES*4);
    exp_sum_kernel<<<(N_EDGES*4+255)/256, 256, 0, stream>>>(alp, ei, mbuf, dbuf, N_EDGES, 4);
    fill_bias_rows<<<(N_NODES*512+255)/256, 256, 0, stream>>>(h1, p1.bias, N_NODES, 512);
    scatter_kernel<4,128><<<EW2, 256, 0, stream>>>(xl1, alp, dbuf, ei, h1, N_EDGES);
    // MeanSubtractionNorm + ReLU
    fill_kernel<<<(NGRAPH*512+32+255)/256, 256, 0, stream>>>(meanb, 0.0f, NGRAPH*512+32);
    seg_sum_kernel<<<(N_NODES+7)/8, 256, 0, stream>>>(h1, batch, meanb, cntb, N_NODES, 512);
    finalize_mean<<<(NGRAPH*512+255)/256, 256, 0, stream>>>(meanb, cntb, NGRAPH, 512);
    sub_mean_relu<<<(N_NODES*512+255)/256, 256, 0, stream>>>(h1, meanb, batch, N_NODES, 512);
    // ---- layer 2: H=1, C=128, in_dim=512 ----
    gemm(stream, h1, p2.Wl, p2.bl, xl1, N_NODES, 512, 128, 0);  // xl2 in xl1
    gemm(stream, h1, p2.Wr, p2.br, xr1, N_NODES, 512, 128, 0);  // xr2 in xr1
    edge_att_kernel<1,128><<<EW2, 256, 0, stream>>>(xl1, xr1, ea, ei,
                                                    p2.We, p2.att, alp, N_EDGES);
    fill_kernel<<<(N_NODES+255)/256, 256, 0, stream>>>(mbuf, -3.0e38f, N_NODES);
    seg_max_kernel<<<(N_EDGES+255)/256, 256, 0, stream>>>(alp, ei, mbuf, N_EDGES, 1);
    fill_kernel<<<(N_NODES+255)/256, 256, 0, stream>>>(dbuf, 0.0f, N_NODES);
    exp_sum_kernel<<<(N_EDGES+255)/256, 256, 0, stream>>>(alp, ei, mbuf, dbuf, N_EDGES, 1);
    fill_bias_rows<<<(N_NODES*128+255)/256, 256, 0, stream>>>(hout, p2.bias, N_NODES, 128);
    scatter_kernel<1,128><<<EW2, 256, 0, stream>>>(xl1, alp, dbuf, ei, hout, N_EDGES);
  };

  auto pool = [&](const float* h, float* gdst) {
    fill_kernel<<<(NGRAPH*128+255)/256, 256, 0, stream>>>(gdst, 0.0f, NGRAPH*128);
    fill_kernel<<<1, 32, 0, stream>>>(cntb, 0.0f, NGRAPH);
    seg_sum_kernel<<<(N_NODES+7)/8, 256, 0, stream>>>(h, batch, gdst, cntb, N_NODES, 128);
    finalize_mean<<<(NGRAPH*128+255)/256, 256, 0, stream>>>(gdst, cntb, NGRAPH, 128);
  };

  auto mlp = [&](const float* X, int rows, const Mlp& m, float* hid, float* o) {
    gemm(stream, X, m.W1, m.b1, hid, rows, 128, 128, 1);         // ReLU hidden
    dot_rows_kernel<<<(rows+7)/8, 256, 0, stream>>>(hid, m.W2, m.b2, o, rows, 128);
  };

  // actor GNN -> h, g
  run_gnn(a1, a2, out_h);
  pool(out_h, out_g);
  // value GNN -> hv, gv, value
  run_gnn(v1, v2, hv);
  pool(hv, gv);
  mlp(gv, NGRAPH, critic, hid32, out_value);
  // source logits over all nodes (reuse hv as hidden), noop over pooled g
  mlp(out_h, N_NODES, srcm, hv, out_src);
  mlp(out_g, NGRAPH, noopm, hid32, out_noop);
}